// GCN_70566312673876
// MI455X (gfx1250) — compile-verified
//
#include <hip/hip_runtime.h>

// -------- WMMA vector types (gfx1250 wave32) --------
typedef __attribute__((ext_vector_type(2))) float v2f;   // 16x4 or 4x16 f32 fragment: 2 VGPRs
typedef __attribute__((ext_vector_type(8))) float v8f;   // 16x16 f32 C/D: 8 VGPRs

#define D 32

// ---------------- degree / norm kernels ----------------
__global__ __launch_bounds__(256) void k_fill_deg(float* __restrict__ deg, int n) {
    int i = blockIdx.x * 256 + threadIdx.x;
    if (i < n) deg[i] = 2.0f;                      // improved=True self-loop weight
}

__global__ __launch_bounds__(256) void k_deg_scatter(const long long* __restrict__ dst,
                                                     const float* __restrict__ w,
                                                     float* __restrict__ deg, int E) {
    int e = blockIdx.x * 256 + threadIdx.x;
    if (e < E) atomicAdd(&deg[(int)dst[e]], w[e]);
}

__global__ __launch_bounds__(256) void k_dinv(float* __restrict__ deg, int n) {
    int i = blockIdx.x * 256 + threadIdx.x;
    if (i < n) {
        float d = deg[i];
        deg[i] = (d > 0.0f) ? rsqrtf(d) : 0.0f;
    }
}

// ---------------- dense transform: H = X @ W via V_WMMA_F32_16X16X4_F32 ----------------
// One wave computes a 16x32 output tile. K=32 consumed in 8 steps of 4.
// A (16x4 f32) layout: lane 0-15 = rows, VGPR0 holds K=0 (lanes 0-15) / K=2 (lanes 16-31),
//                      VGPR1 holds K=1 / K=3.   B (4x16) mirrors with N across lanes.
// C/D (16x16 f32): VGPR r = row r for lanes 0-15, row r+8 for lanes 16-31.
__global__ __launch_bounds__(256) void k_gemm(const float* __restrict__ X,
                                              const float* __restrict__ W,
                                              float* __restrict__ H, int n) {
    int wave = (int)((blockIdx.x * (unsigned)blockDim.x + threadIdx.x) >> 5);
    int lane = threadIdx.x & 31;
    int row0 = wave * 16;
    if (row0 >= n) return;                        // wave-uniform: EXEC stays all-ones
    int half = lane >> 4;                         // 0: lanes 0-15, 1: lanes 16-31
    int lm   = lane & 15;

    int arow = row0 + lm;
    if (arow >= n) arow = n - 1;                  // clamp loads; stores are guarded below
    const float* xr = X + (size_t)arow * D;

    v8f c0 = {};                                  // output cols 0..15
    v8f c1 = {};                                  // output cols 16..31
#pragma unroll
    for (int kk = 0; kk < 8; ++kk) {
        int k0 = kk * 4 + half * 2;
        v2f a;                                    // A fragment: X[row, k0], X[row, k0+1]
        a.x = xr[k0];
        a.y = xr[k0 + 1];
        v2f b0, b1;                               // B fragments: W[k, n]
        b0.x = W[k0 * D + lm];
        b0.y = W[(k0 + 1) * D + lm];
        b1.x = W[k0 * D + 16 + lm];
        b1.y = W[(k0 + 1) * D + 16 + lm];
        c0 = __builtin_amdgcn_wmma_f32_16x16x4_f32(false, a, false, b0, (short)0, c0, false, false);
        c1 = __builtin_amdgcn_wmma_f32_16x16x4_f32(false, a, false, b1, (short)0, c1, false, false);
    }

#pragma unroll
    for (int r = 0; r < 8; ++r) {
        int m = row0 + r + 8 * half;
        if (m < n) {
            H[(size_t)m * D + lm]      = c0[r];
            H[(size_t)m * D + 16 + lm] = c1[r];
        }
    }
}

// ---------------- aggregation ----------------
// AGG[i,f] = 2*dinv[i]^2 * H[i,f] + b[f]   (self-loop term + bias)
__global__ __launch_bounds__(256) void k_agg_init(const float* __restrict__ H,
                                                  const float* __restrict__ dinv,
                                                  const float* __restrict__ b,
                                                  float* __restrict__ AGG, int nf) {
    int i = blockIdx.x * 256 + threadIdx.x;
    if (i < nf) {
        int node = i >> 5;
        int f    = i & 31;
        float di = dinv[node];
        AGG[i] = 2.0f * di * di * H[i] + b[f];
    }
}

// One wave per edge, one lane per feature: coalesced 128B gather of H[src,:],
// 32 f32 atomics into AGG[dst,:].
__global__ __launch_bounds__(256) void k_edge_scatter(const long long* __restrict__ src,
                                                      const long long* __restrict__ dst,
                                                      const float* __restrict__ w,
                                                      const float* __restrict__ dinv,
                                                      const float* __restrict__ H,
                                                      float* __restrict__ AGG, int E) {
    int e    = (int)((blockIdx.x * (unsigned)blockDim.x + threadIdx.x) >> 5);
    int lane = threadIdx.x & 31;
    if (e >= E) return;
    int s = (int)src[e];
    int d = (int)dst[e];
    float coef = dinv[s] * w[e] * dinv[d];
    float v = coef * H[(size_t)s * D + lane];
    atomicAdd(&AGG[(size_t)d * D + lane], v);
}

__global__ __launch_bounds__(256) void k_relu(const float* __restrict__ in,
                                              float* __restrict__ out, int nf) {
    int i = blockIdx.x * 256 + threadIdx.x;
    if (i < nf) out[i] = fmaxf(in[i], 0.0f);
}

// ---------------- driver ----------------
extern "C" void kernel_launch(void* const* d_in, const int* in_sizes, int n_in,
                              void* d_out, int out_size, void* d_ws, size_t ws_size,
                              hipStream_t stream) {
    const float*     x  = (const float*)d_in[0];
    const long long* ei = (const long long*)d_in[1];   // int64 edge_index [2,E]
    const float*     ew = (const float*)d_in[2];
    const float*     W1 = (const float*)d_in[3];
    const float*     b1 = (const float*)d_in[4];
    const float*     W2 = (const float*)d_in[5];
    const float*     b2 = (const float*)d_in[6];
    float* out = (float*)d_out;

    int N = in_sizes[0] / D;
    int E = in_sizes[2];
    const long long* src = ei;
    const long long* dst = ei + E;

    float* ws   = (float*)d_ws;
    float* dinv = ws;                         // N floats (deg, then dinv in place)
    float* B1   = ws + N;                     // N*D floats
    float* B2   = B1 + (size_t)N * D;         // N*D floats

    int NF    = N * D;
    int gN    = (N  + 255) / 256;
    int gE    = (E  + 255) / 256;
    int gNF   = (NF + 255) / 256;
    int gGemm = (N + 127) / 128;              // 8 waves/block * 16 rows/wave
    int gEdge = (E + 7) / 8;                  // 8 edges per 256-thread block

    // normalization
    k_fill_deg   <<<gN,    256, 0, stream>>>(dinv, N);
    k_deg_scatter<<<gE,    256, 0, stream>>>(dst, ew, dinv, E);
    k_dinv       <<<gN,    256, 0, stream>>>(dinv, N);

    // layer 1: h1 = relu(agg(x@W1) + b1)
    k_gemm        <<<gGemm, 256, 0, stream>>>(x, W1, B1, N);
    k_agg_init    <<<gNF,   256, 0, stream>>>(B1, dinv, b1, B2, NF);
    k_edge_scatter<<<gEdge, 256, 0, stream>>>(src, dst, ew, dinv, B1, B2, E);
    k_relu        <<<gNF,   256, 0, stream>>>(B2, B1, NF);     // B1 = h1

    // layer 2: out = relu(agg(h1@W2) + b2)
    k_gemm        <<<gGemm, 256, 0, stream>>>(B1, W2, B2, N);
    k_agg_init    <<<gNF,   256, 0, stream>>>(B2, dinv, b2, B1, NF);
    k_edge_scatter<<<gEdge, 256, 0, stream>>>(src, dst, ew, dinv, B2, B1, E);
    k_relu        <<<gNF,   256, 0, stream>>>(B1, out, NF);
}